// GNNModel_25512105738587
// MI455X (gfx1250) — compile-verified
//
#include <hip/hip_runtime.h>

namespace {

constexpr int NN   = 100000;   // nodes
constexpr int NE   = 600000;   // edges
constexpr int NG   = 128;      // graphs
constexpr int C    = 128;      // channels
constexpr int C2   = 256;      // 2*C
constexpr int FH   = 512;      // fc hidden
constexpr int NCLS = 250;      // classes
constexpr float EPS_MSG = 1e-7f;
constexpr float EPS_BN  = 1e-5f;
constexpr float EPS_SM  = 1e-16f;

typedef float v2f __attribute__((ext_vector_type(2)));
typedef float v8f __attribute__((ext_vector_type(8)));

__device__ __forceinline__ v8f wmma4(v2f a, v2f b, v8f c) {
  // V_WMMA_F32_16X16X4_F32 : D(16x16,f32) = A(16x4,f32) * B(4x16,f32) + C
  return __builtin_amdgcn_wmma_f32_16x16x4_f32(
      /*neg_a=*/false, a, /*neg_b=*/false, b,
      /*c_mod=*/(short)0, c, /*reuse_a=*/false, /*reuse_b=*/false);
}

// ---------------- edge pass 1: per-(dst,channel) max of msg -----------------
// msg = relu(v_emb[x[src]] + e_emb[edge_attr]) + EPS_MSG  (always > 0)
// positive floats order like ints -> exact atomicMax on bit pattern, init 0.
__global__ __launch_bounds__(256) void edge_max_kernel(
    const int* __restrict__ ei, const int* __restrict__ eattr,
    const int* __restrict__ x, const float* __restrict__ v_emb,
    const float* __restrict__ e_emb, float* __restrict__ mmax) {
  int gid = blockIdx.x * 256 + threadIdx.x;
  int e = gid >> 5;
  if (e >= NE) return;
  int c = (gid & 31) << 2;
  int src = ei[e], dst = ei[NE + e];
  int xs = x[src], ea = eattr[e];
  float4 hv = *(const float4*)(v_emb + xs * C + c);
  float4 ev = *(const float4*)(e_emb + ea * C + c);
  float m0 = fmaxf(hv.x + ev.x, 0.f) + EPS_MSG;
  float m1 = fmaxf(hv.y + ev.y, 0.f) + EPS_MSG;
  float m2 = fmaxf(hv.z + ev.z, 0.f) + EPS_MSG;
  float m3 = fmaxf(hv.w + ev.w, 0.f) + EPS_MSG;
  int* mm = (int*)(mmax + dst * C + c);
  atomicMax(mm + 0, __float_as_int(m0));
  atomicMax(mm + 1, __float_as_int(m1));
  atomicMax(mm + 2, __float_as_int(m2));
  atomicMax(mm + 3, __float_as_int(m3));
}

// -------- edge pass 2: denom += exp(msg-mmax); numer += msg*exp(...) --------
__global__ __launch_bounds__(256) void edge_sum_kernel(
    const int* __restrict__ ei, const int* __restrict__ eattr,
    const int* __restrict__ x, const float* __restrict__ v_emb,
    const float* __restrict__ e_emb, const float* __restrict__ mmax,
    float* __restrict__ denom, float* __restrict__ numer) {
  int gid = blockIdx.x * 256 + threadIdx.x;
  int e = gid >> 5;
  if (e >= NE) return;
  int c = (gid & 31) << 2;
  int src = ei[e], dst = ei[NE + e];
  int xs = x[src], ea = eattr[e];
  float4 hv = *(const float4*)(v_emb + xs * C + c);
  float4 ev = *(const float4*)(e_emb + ea * C + c);
  float4 mx = *(const float4*)(mmax + dst * C + c);
  float m0 = fmaxf(hv.x + ev.x, 0.f) + EPS_MSG;
  float m1 = fmaxf(hv.y + ev.y, 0.f) + EPS_MSG;
  float m2 = fmaxf(hv.z + ev.z, 0.f) + EPS_MSG;
  float m3 = fmaxf(hv.w + ev.w, 0.f) + EPS_MSG;
  float w0 = __expf(m0 - mx.x);
  float w1 = __expf(m1 - mx.y);
  float w2 = __expf(m2 - mx.z);
  float w3 = __expf(m3 - mx.w);
  float* dn = denom + dst * C + c;
  float* nu = numer + dst * C + c;
  atomicAdd(dn + 0, w0);  atomicAdd(nu + 0, m0 * w0);
  atomicAdd(dn + 1, w1);  atomicAdd(nu + 1, m1 * w1);
  atomicAdd(dn + 2, w2);  atomicAdd(nu + 2, m2 * w2);
  atomicAdd(dn + 3, w3);  atomicAdd(nu + 3, m3 * w3);
}

// ------------- out = numer/(denom+eps) + h   (in place over numer) ----------
__global__ __launch_bounds__(256) void node_finalize_kernel(
    const int* __restrict__ x, const float* __restrict__ v_emb,
    const float* __restrict__ denom, float* __restrict__ numer) {
  int gid = blockIdx.x * 256 + threadIdx.x;
  int n = gid >> 5;
  if (n >= NN) return;
  int c = (gid & 31) << 2;
  float4 hv = *(const float4*)(v_emb + x[n] * C + c);
  float4 nu = *(const float4*)(numer + n * C + c);
  float4 dn = *(const float4*)(denom + n * C + c);
  float4 o;
  o.x = nu.x / (dn.x + EPS_SM) + hv.x;
  o.y = nu.y / (dn.y + EPS_SM) + hv.y;
  o.z = nu.z / (dn.z + EPS_SM) + hv.z;
  o.w = nu.w / (dn.w + EPS_SM) + hv.w;
  *(float4*)(numer + n * C + c) = o;
}

// ---------------- GEMM1: Z[N,256] = out[N,128] @ w1[128,256] + b1 -----------
// 8 waves/block; block owns 16 rows; wave w owns cols [32w,32w+32) (2 tiles).
__global__ __launch_bounds__(256) void gemm1_kernel(
    const float* __restrict__ A, const float* __restrict__ B,
    const float* __restrict__ bias, float* __restrict__ Z) {
  int lane = threadIdx.x & 31;
  int wave = threadIdx.x >> 5;
  int row0 = blockIdx.x * 16;
  int col0 = wave * 32;
  int r  = row0 + (lane & 15);
  int kh = (lane >> 4) << 1;            // 0 or 2
  int cb = lane & 15;
  v8f acc0 = {}, acc1 = {};
  for (int k = 0; k < C; k += 4) {
    int ka = k + kh;
    v2f a; a.x = A[r * C + ka];               a.y = A[r * C + ka + 1];
    v2f b0; b0.x = B[ka * C2 + col0 + cb];    b0.y = B[(ka + 1) * C2 + col0 + cb];
    v2f b1; b1.x = B[ka * C2 + col0 + 16 + cb]; b1.y = B[(ka + 1) * C2 + col0 + 16 + cb];
    acc0 = wmma4(a, b0, acc0);
    acc1 = wmma4(a, b1, acc1);
  }
  int rb = row0 + ((lane >> 4) << 3);
  int cA = col0 + cb, cB = col0 + 16 + cb;
  float bA = bias[cA], bB = bias[cB];
  #pragma unroll
  for (int v = 0; v < 8; ++v) {
    Z[(rb + v) * C2 + cA] = acc0[v] + bA;
    Z[(rb + v) * C2 + cB] = acc1[v] + bB;
  }
}

// ---------------- BN stats: per-channel sum / sumsq in f64 ------------------
__global__ __launch_bounds__(256) void bn_stats_kernel(
    const float* __restrict__ Z, double* __restrict__ gsum,
    double* __restrict__ gsumsq) {
  int c = threadIdx.x;                  // channel 0..255
  int r0 = blockIdx.x * 256;
  int r1 = min(r0 + 256, NN);
  double s = 0.0, sq = 0.0;
  for (int r = r0; r < r1; ++r) {
    float z = Z[r * C2 + c];
    s += (double)z;
    sq += (double)z * (double)z;
  }
  atomicAdd(&gsum[c], s);
  atomicAdd(&gsumsq[c], sq);
}

__global__ void bn_params_kernel(const double* __restrict__ gsum,
                                 const double* __restrict__ gsumsq,
                                 const float* __restrict__ gamma,
                                 const float* __restrict__ beta,
                                 float* __restrict__ bnS, float* __restrict__ bnT) {
  int c = threadIdx.x;
  double mu  = gsum[c] / (double)NN;
  double var = gsumsq[c] / (double)NN - mu * mu;
  float inv = rsqrtf((float)var + EPS_BN);
  float sc = gamma[c] * inv;
  bnS[c] = sc;
  bnT[c] = beta[c] - (float)mu * sc;
}

// ------- GEMM2: node_out[N,128] = relu(bn(Z))[N,256] @ w2[256,128] + b2 -----
// BN+ReLU fused into A-fragment load. 8 waves, 1 tile (16 cols) each.
__global__ __launch_bounds__(256) void gemm2_kernel(
    const float* __restrict__ Z, const float* __restrict__ B,
    const float* __restrict__ bias, const float* __restrict__ bnS,
    const float* __restrict__ bnT, float* __restrict__ O) {
  int lane = threadIdx.x & 31;
  int wave = threadIdx.x >> 5;
  int row0 = blockIdx.x * 16;
  int col0 = wave * 16;
  int r  = row0 + (lane & 15);
  int kh = (lane >> 4) << 1;
  int cb = lane & 15;
  v8f acc = {};
  for (int k = 0; k < C2; k += 4) {
    int ka = k + kh;
    float z0 = Z[r * C2 + ka], z1 = Z[r * C2 + ka + 1];
    v2f a;
    a.x = fmaxf(z0 * bnS[ka] + bnT[ka], 0.f);
    a.y = fmaxf(z1 * bnS[ka + 1] + bnT[ka + 1], 0.f);
    v2f b; b.x = B[ka * C + col0 + cb]; b.y = B[(ka + 1) * C + col0 + cb];
    acc = wmma4(a, b, acc);
  }
  int rb = row0 + ((lane >> 4) << 3);
  int cc = col0 + cb;
  float bb = bias[cc];
  #pragma unroll
  for (int v = 0; v < 8; ++v) O[(rb + v) * C + cc] = acc[v] + bb;
}

// ------------------------------ mean pool ----------------------------------
__global__ __launch_bounds__(256) void pool_cnt_kernel(
    const int* __restrict__ batch, float* __restrict__ cnt) {
  int n = blockIdx.x * 256 + threadIdx.x;
  if (n < NN) atomicAdd(&cnt[batch[n]], 1.0f);
}

__global__ __launch_bounds__(256) void pool_sum_kernel(
    const float* __restrict__ node_out, const int* __restrict__ batch,
    float* __restrict__ pooled) {
  int gid = blockIdx.x * 256 + threadIdx.x;
  int n = gid >> 5;
  if (n >= NN) return;
  int c = (gid & 31) << 2;
  int g = batch[n];
  float4 v = *(const float4*)(node_out + n * C + c);
  float* p = pooled + g * C + c;
  atomicAdd(p + 0, v.x);
  atomicAdd(p + 1, v.y);
  atomicAdd(p + 2, v.z);
  atomicAdd(p + 3, v.w);
}

__global__ __launch_bounds__(256) void pool_div_kernel(
    float* __restrict__ pooled, const float* __restrict__ cnt) {
  int gid = blockIdx.x * 256 + threadIdx.x;   // NG*C = 16384
  int g = gid >> 7;
  pooled[gid] /= fmaxf(cnt[g], 1.0f);
}

// ------------- FC1: hdn[128,512] = relu(pooled[128,128]@fc1w + b) -----------
__global__ __launch_bounds__(256) void fc1_kernel(
    const float* __restrict__ A, const float* __restrict__ B,
    const float* __restrict__ bias, float* __restrict__ O) {
  int lane = threadIdx.x & 31;
  int wave = threadIdx.x >> 5;
  int row0 = blockIdx.x * 16;
  int col0 = blockIdx.y * 256 + wave * 32;
  int r  = row0 + (lane & 15);
  int kh = (lane >> 4) << 1;
  int cb = lane & 15;
  v8f acc0 = {}, acc1 = {};
  for (int k = 0; k < C; k += 4) {
    int ka = k + kh;
    v2f a; a.x = A[r * C + ka];               a.y = A[r * C + ka + 1];
    v2f b0; b0.x = B[ka * FH + col0 + cb];    b0.y = B[(ka + 1) * FH + col0 + cb];
    v2f b1; b1.x = B[ka * FH + col0 + 16 + cb]; b1.y = B[(ka + 1) * FH + col0 + 16 + cb];
    acc0 = wmma4(a, b0, acc0);
    acc1 = wmma4(a, b1, acc1);
  }
  int rb = row0 + ((lane >> 4) << 3);
  int cA = col0 + cb, cB = col0 + 16 + cb;
  float bA = bias[cA], bB = bias[cB];
  #pragma unroll
  for (int v = 0; v < 8; ++v) {
    O[(rb + v) * FH + cA] = fmaxf(acc0[v] + bA, 0.f);
    O[(rb + v) * FH + cB] = fmaxf(acc1[v] + bB, 0.f);
  }
}

// ---------- FC2: out[128,250] = hdn[128,512] @ fc2w[512,250] + b ------------
// N padded to 256; B loads use select (no EXEC change), stores guarded.
__global__ __launch_bounds__(256) void fc2_kernel(
    const float* __restrict__ A, const float* __restrict__ B,
    const float* __restrict__ bias, float* __restrict__ O) {
  int lane = threadIdx.x & 31;
  int wave = threadIdx.x >> 5;
  int row0 = blockIdx.x * 16;
  int col0 = wave * 32;
  int r  = row0 + (lane & 15);
  int kh = (lane >> 4) << 1;
  int cb = lane & 15;
  int cA = col0 + cb, cB = col0 + 16 + cb;
  v8f acc0 = {}, acc1 = {};
  for (int k = 0; k < FH; k += 4) {
    int ka = k + kh;
    v2f a; a.x = A[r * FH + ka]; a.y = A[r * FH + ka + 1];
    v2f b0, b1;
    b0.x = (cA < NCLS) ? B[ka * NCLS + cA] : 0.f;
    b0.y = (cA < NCLS) ? B[(ka + 1) * NCLS + cA] : 0.f;
    b1.x = (cB < NCLS) ? B[ka * NCLS + cB] : 0.f;
    b1.y = (cB < NCLS) ? B[(ka + 1) * NCLS + cB] : 0.f;
    acc0 = wmma4(a, b0, acc0);
    acc1 = wmma4(a, b1, acc1);
  }
  int rb = row0 + ((lane >> 4) << 3);
  float bA = (cA < NCLS) ? bias[cA] : 0.f;
  float bB = (cB < NCLS) ? bias[cB] : 0.f;
  #pragma unroll
  for (int v = 0; v < 8; ++v) {
    if (cA < NCLS) O[(rb + v) * NCLS + cA] = acc0[v] + bA;
    if (cB < NCLS) O[(rb + v) * NCLS + cB] = acc1[v] + bB;
  }
}

} // anonymous namespace

extern "C" void kernel_launch(void* const* d_in, const int* in_sizes, int n_in,
                              void* d_out, int out_size, void* d_ws, size_t ws_size,
                              hipStream_t stream) {
  const int*   x     = (const int*)d_in[0];
  const int*   ei    = (const int*)d_in[1];   // [2,E] row-major
  const int*   eattr = (const int*)d_in[2];
  const int*   batch = (const int*)d_in[3];
  const float* v_emb = (const float*)d_in[4];
  const float* e_emb = (const float*)d_in[5];
  const float* w1    = (const float*)d_in[6];
  const float* b1    = (const float*)d_in[7];
  const float* gamma = (const float*)d_in[8];
  const float* beta  = (const float*)d_in[9];
  const float* w2    = (const float*)d_in[10];
  const float* b2    = (const float*)d_in[11];
  const float* fc1w  = (const float*)d_in[12];
  const float* fc1b  = (const float*)d_in[13];
  const float* fc2w  = (const float*)d_in[14];
  const float* fc2b  = (const float*)d_in[15];
  float* out = (float*)d_out;

  const size_t NC = (size_t)NN * C;
  float* base  = (float*)d_ws;
  float* mmax  = base;            // [N,C]   -> later low half of Z
  float* denom = base + NC;       // [N,C]   -> later high half of Z
  float* numer = base + 2 * NC;   // [N,C]   -> out -> node_out
  float* zbuf  = base;            // [N,2C] overlays mmax+denom (dead by then)
  float*  smallr = base + 3 * NC;
  double* dsum   = (double*)smallr;       // 256 f64
  double* dsumsq = dsum + C2;             // 256 f64
  float* bnS    = (float*)(dsumsq + C2);  // 256
  float* bnT    = bnS + C2;               // 256
  float* cnt    = bnT + C2;               // 128
  float* pooled = cnt + NG;               // 128*128
  float* hdn    = pooled + (size_t)NG * C;// 128*512

  size_t zero_bytes = 3 * NC * sizeof(float) + 2 * C2 * sizeof(double) +
                      (size_t)(2 * C2 + NG + NG * C) * sizeof(float);
  hipMemsetAsync(d_ws, 0, zero_bytes, stream);

  int eb = (NE * 32 + 255) / 256;   // edge kernels: 32 threads/edge
  int nb = (NN * 32 + 255) / 256;   // node kernels: 32 threads/node
  int sb = (NN + 255) / 256;

  edge_max_kernel<<<eb, 256, 0, stream>>>(ei, eattr, x, v_emb, e_emb, mmax);
  edge_sum_kernel<<<eb, 256, 0, stream>>>(ei, eattr, x, v_emb, e_emb, mmax, denom, numer);
  node_finalize_kernel<<<nb, 256, 0, stream>>>(x, v_emb, denom, numer);

  gemm1_kernel<<<NN / 16, 256, 0, stream>>>(numer, w1, b1, zbuf);
  bn_stats_kernel<<<sb, 256, 0, stream>>>(zbuf, dsum, dsumsq);
  bn_params_kernel<<<1, 256, 0, stream>>>(dsum, dsumsq, gamma, beta, bnS, bnT);
  gemm2_kernel<<<NN / 16, 256, 0, stream>>>(zbuf, w2, b2, bnS, bnT, numer);

  pool_cnt_kernel<<<sb, 256, 0, stream>>>(batch, cnt);
  pool_sum_kernel<<<nb, 256, 0, stream>>>(numer, batch, pooled);
  pool_div_kernel<<<(NG * C) / 256, 256, 0, stream>>>(pooled, cnt);

  fc1_kernel<<<dim3(NG / 16, FH / 256), 256, 0, stream>>>(pooled, fc1w, fc1b, hdn);
  fc2_kernel<<<NG / 16, 256, 0, stream>>>(hdn, fc2w, fc2b, out);
}